// AmplitudeEncoder_33603824124046
// MI455X (gfx1250) — compile-verified
//
#include <hip/hip_runtime.h>

// Problem constants (AmplitudeEncoder: 64 x 784 -> 64 x 1001 x 1001 outer products)
#define DDIM 1001      // statevector dimension
#define FDIM 784       // feature count
#define BATCH 64
#define TILE 16
#define NT 63          // ceil(1001/16) tiles per dim; 63*16 = 1008
#define WSROW 1008     // zero-padded state length per sample in workspace

typedef __attribute__((ext_vector_type(2))) float v2f;
typedef __attribute__((ext_vector_type(8))) float v8f;

// ---------------------------------------------------------------------------
// Kernel 1: per-sample L2 norm, write normalized + zero-padded state to ws.
// 64 blocks (one per sample) x 256 threads (8 wave32s).
// ---------------------------------------------------------------------------
__global__ void normalize_state(const float* __restrict__ x, float* __restrict__ ws) {
    __shared__ float red[256];
    const int b   = blockIdx.x;
    const int tid = threadIdx.x;
    const float* xb = x + b * FDIM;

    float s = 0.0f;
    for (int i = tid; i < FDIM; i += 256) {
        float v = xb[i];
        s += v * v;
    }
    red[tid] = s;
    __syncthreads();
    for (int ofs = 128; ofs > 0; ofs >>= 1) {
        if (tid < ofs) red[tid] += red[tid + ofs];
        __syncthreads();
    }
    const float inv = 1.0f / sqrtf(red[0]);

    float* wb = ws + b * WSROW;
    for (int i = tid; i < WSROW; i += 256) {
        wb[i] = (i < FDIM) ? xb[i] * inv : 0.0f;
    }
}

// ---------------------------------------------------------------------------
// Kernel 2: one wave32 per 16x16 output tile, rank-1 product via
// V_WMMA_F32_16X16X4_F32 (only K=0 populated -> exact f32 multiply).
// 8 waves per 256-thread block.
// ---------------------------------------------------------------------------
__global__ void outer_wmma(const float* __restrict__ ws, float* __restrict__ out) {
    const int lane = threadIdx.x & 31;
    const int wave = threadIdx.x >> 5;
    const int w    = blockIdx.x * 8 + wave;            // global wave id

    const int b  = w / (NT * NT);
    const int r  = w - b * (NT * NT);
    const int ti = r / NT;
    const int tj = r - ti * NT;
    const int i0 = ti * TILE;
    const int j0 = tj * TILE;

    const float* sb = ws + b * WSROW;
    const int half = lane >> 4;        // 0: lanes 0-15, 1: lanes 16-31
    const int n    = lane & 15;

    // A (16x4 f32): VGPR0 lanes 0-15 hold K=0 (M=lane), lanes 16-31 hold K=2.
    // B (4x16 f32): VGPR0 lanes 0-15 hold K=0 row (N=lane).
    // Zero everything except K=0 -> D[m][n] = s_i[m] * s_j[n] exactly.
    const float av = sb[i0 + n];
    const float bv = sb[j0 + n];
    v2f a;  a.x  = half ? 0.0f : av;  a.y  = 0.0f;
    v2f bm; bm.x = half ? 0.0f : bv;  bm.y = 0.0f;
    v8f c = {};
    c = __builtin_amdgcn_wmma_f32_16x16x4_f32(
            /*neg_a=*/false, a, /*neg_b=*/false, bm,
            /*c_mod=*/(short)0, c, /*reuse_a=*/false, /*reuse_b=*/false);

    // C/D layout: VGPR v, lane l -> row = i0 + 8*(l>=16) + v, col = j0 + (l&15)
    const int row0 = i0 + half * 8;
    const int col  = j0 + n;
    float* outp = out + (size_t)b * DDIM * DDIM + (size_t)row0 * DDIM + col;

    if (i0 + TILE <= DDIM && j0 + TILE <= DDIM) {
        // Interior tile: 8 coalesced non-temporal b32 stores (streaming 256 MB).
#pragma unroll
        for (int v = 0; v < 8; ++v) {
            __builtin_nontemporal_store(c[v], outp + (size_t)v * DDIM);
        }
    } else {
        // Boundary tile (1008 > 1001): guard rows/cols.
        if (col < DDIM) {
#pragma unroll
            for (int v = 0; v < 8; ++v) {
                if (row0 + v < DDIM) {
                    __builtin_nontemporal_store(c[v], outp + (size_t)v * DDIM);
                }
            }
        }
    }
}

// ---------------------------------------------------------------------------
extern "C" void kernel_launch(void* const* d_in, const int* in_sizes, int n_in,
                              void* d_out, int out_size, void* d_ws, size_t ws_size,
                              hipStream_t stream) {
    (void)in_sizes; (void)n_in; (void)out_size; (void)ws_size;

    const float* x = (const float*)d_in[0];   // [64, 784] float32
    float* out = (float*)d_out;               // [64, 1001, 1001] float32
    float* ws  = (float*)d_ws;                // 64 * 1008 floats = 258 KB

    normalize_state<<<BATCH, 256, 0, stream>>>(x, ws);

    const int total_waves = BATCH * NT * NT;  // 254016 tiles, one wave each
    outer_wmma<<<total_waves / 8, 256, 0, stream>>>(ws, out);
}